// DKVMN_88751204204927
// MI455X (gfx1250) — compile-verified
//
#include <hip/hip_runtime.h>

typedef __attribute__((ext_vector_type(2))) float v2f;
typedef __attribute__((ext_vector_type(8))) float v8f;

#define B_   512
#define T_   200
#define BT_  (B_ * T_)
#define M_   50
#define DK_  64
#define DV_  64
#define H_   128
#define NSK_ 1000

static __device__ __forceinline__ v8f wmma_f32_k4(v2f a, v2f b, v8f c) {
  // V_WMMA_F32_16X16X4_F32 : exact fp32 matrix-FMA, 16x16 tile, K=4
  return __builtin_amdgcn_wmma_f32_16x16x4_f32(false, a, false, b, (short)0, c,
                                               false, false);
}

// ---------------------------------------------------------------------------
// Kernel 1: per-(b,t) precompute.
//   q = skill_embed[s];  w = softmax(q @ K^T)  (padded to 64, zeros m>=50)
//   v = interaction_embed[s + c*N];  e = sigmoid(v@We+be);  a = tanh(v@Wa+ba)
// One wave owns a 16-row tile; 8 waves (256 thr) per block -> 128 rows/block.
// ---------------------------------------------------------------------------
__global__ __launch_bounds__(256) void dkvmn_precompute(
    const int* __restrict__ skill_seq, const int* __restrict__ correct_seq,
    const float* __restrict__ skill_embed, const float* __restrict__ key_memory,
    const float* __restrict__ interaction_embed,
    const float* __restrict__ erase_W, const float* __restrict__ erase_b,
    const float* __restrict__ add_W, const float* __restrict__ add_b,
    float* __restrict__ w_buf, float* __restrict__ e_buf,
    float* __restrict__ a_buf) {
  __shared__ float qtile[8][16][DK_];  // 32 KB (reused for logits)
  __shared__ float vtile[8][16][DV_];  // 32 KB

  const int tid = threadIdx.x;
  const int wave = tid >> 5;
  const int lane = tid & 31;
  const int r0 = (blockIdx.x * 8 + wave) * 16;

  {  // gather q, v into LDS (each lane: half a row = 32 floats)
    int row = lane >> 1, half = (lane & 1) * 32;
    int g = r0 + row;
    int s = skill_seq[g];
    int c = correct_seq[g];
    const float4* qsrc = (const float4*)(skill_embed + (size_t)s * DK_ + half);
    const float4* vsrc =
        (const float4*)(interaction_embed + (size_t)(s + c * NSK_) * DV_ + half);
    float4* qd = (float4*)&qtile[wave][row][half];
    float4* vd = (float4*)&vtile[wave][row][half];
#pragma unroll
    for (int j = 0; j < 8; ++j) {
      qd[j] = qsrc[j];
      vd[j] = vsrc[j];
    }
  }
  __syncthreads();

  const int mrow = lane & 15;       // A row / B column sub-index
  const int kh = (lane >> 4) << 1;  // K sub-offset per ISA 16x4 A layout
  const int dcol = lane & 15;       // D column sub-index
  const int rbase = (lane < 16) ? 0 : 8;

  // ---- logits = q @ key_memory^T ----
  // N is padded 50->64; columns n>=50 are never consumed downstream (softmax
  // only reads m<50 and w_buf is explicitly zero-padded), so we just clamp the
  // row index to stay in-bounds and let garbage land in the ignored columns.
  // This keeps every load unconditional -> no exec-mask divergence in the
  // WMMA loop.
  v8f lacc[4];
#pragma unroll
  for (int nt = 0; nt < 4; ++nt) lacc[nt] = (v8f)0.0f;

  const float* kmrow[4];
#pragma unroll
  for (int nt = 0; nt < 4; ++nt) {
    int n = nt * 16 + mrow;
    int nc = (n < M_) ? n : (M_ - 1);  // clamp: keeps loads in-bounds
    kmrow[nt] = key_memory + (size_t)nc * DK_;
  }

  for (int kk = 0; kk < 16; ++kk) {
    int k0 = kk * 4;
    v2f av;
    av.x = qtile[wave][mrow][k0 + kh];
    av.y = qtile[wave][mrow][k0 + kh + 1];
#pragma unroll
    for (int nt = 0; nt < 4; ++nt) {
      v2f bv;  // B[k][n] = key_memory[n][k]
      bv.x = kmrow[nt][k0 + kh];
      bv.y = kmrow[nt][k0 + kh + 1];
      lacc[nt] = wmma_f32_k4(av, bv, lacc[nt]);
    }
  }
  __syncthreads();
  // stash logits into qtile (q no longer needed)
#pragma unroll
  for (int nt = 0; nt < 4; ++nt)
#pragma unroll
    for (int j = 0; j < 8; ++j)
      qtile[wave][rbase + j][nt * 16 + dcol] = lacc[nt][j];
  __syncthreads();

  // ---- softmax over m=0..49, one row per lane 0..15 ----
  if (lane < 16) {
    float* lg = qtile[wave][lane];
    float mx = -1e30f;
    for (int m = 0; m < M_; ++m) mx = fmaxf(mx, lg[m]);
    float sum = 0.0f;
    for (int m = 0; m < M_; ++m) {
      float ex = __expf(lg[m] - mx);
      lg[m] = ex;
      sum += ex;
    }
    float inv = 1.0f / sum;
    float* dst = w_buf + (size_t)(r0 + lane) * 64;
    for (int m = 0; m < M_; ++m) dst[m] = lg[m] * inv;
    for (int m = M_; m < 64; ++m) dst[m] = 0.0f;  // pad for scan kernel
  }

  // ---- e = sigmoid(v@We+be), a = tanh(v@Wa+ba) ----
  v8f eacc[4], aacc[4];
#pragma unroll
  for (int nt = 0; nt < 4; ++nt) {
    eacc[nt] = (v8f)0.0f;
    aacc[nt] = (v8f)0.0f;
  }
  for (int kk = 0; kk < 16; ++kk) {
    int k0 = kk * 4;
    v2f av;
    av.x = vtile[wave][mrow][k0 + kh];
    av.y = vtile[wave][mrow][k0 + kh + 1];
#pragma unroll
    for (int nt = 0; nt < 4; ++nt) {
      int n = nt * 16 + mrow;
      v2f be, ba;
      be.x = erase_W[(size_t)(k0 + kh) * DV_ + n];
      be.y = erase_W[(size_t)(k0 + kh + 1) * DV_ + n];
      ba.x = add_W[(size_t)(k0 + kh) * DV_ + n];
      ba.y = add_W[(size_t)(k0 + kh + 1) * DV_ + n];
      eacc[nt] = wmma_f32_k4(av, be, eacc[nt]);
      aacc[nt] = wmma_f32_k4(av, ba, aacc[nt]);
    }
  }
#pragma unroll
  for (int nt = 0; nt < 4; ++nt) {
    int col = nt * 16 + dcol;
    float eb = erase_b[col];
    float ab = add_b[col];
#pragma unroll
    for (int j = 0; j < 8; ++j) {
      int g = r0 + rbase + j;
      float ev = 1.0f / (1.0f + __expf(-(eacc[nt][j] + eb)));
      float avv = tanhf(aacc[nt][j] + ab);
      e_buf[(size_t)g * 64 + col] = ev;
      a_buf[(size_t)g * 64 + col] = avv;
    }
  }
}

// ---------------------------------------------------------------------------
// Kernel 2: sequential scan over T.  One block per batch element; memory state
// (50x64 f32 = 12.8 KB) lives in LDS for all 200 steps.
// ---------------------------------------------------------------------------
__global__ __launch_bounds__(256) void dkvmn_scan(
    const float* __restrict__ value_init, const float* __restrict__ w_buf,
    const float* __restrict__ e_buf, const float* __restrict__ a_buf,
    float* __restrict__ reads_buf) {
  __shared__ float mem[M_][DV_];
  __shared__ float wv[M_];
  __shared__ float ev[DV_];
  __shared__ float av[DV_];

  const int tid = threadIdx.x;
  const int b = blockIdx.x;

  for (int i = tid; i < M_ * DV_; i += 256) mem[i >> 6][i & 63] = value_init[i];
  __syncthreads();

  for (int t = 0; t < T_; ++t) {
    size_t base = ((size_t)b * T_ + t) * 64;
    if (tid < M_)
      wv[tid] = w_buf[base + tid];
    else if (tid >= 64 && tid < 128)
      ev[tid - 64] = e_buf[base + (tid - 64)];
    else if (tid >= 128 && tid < 192)
      av[tid - 128] = a_buf[base + (tid - 128)];
    __syncthreads();

    if (tid < DV_) {  // read_t = w_t @ mem (pre-update)
      float s = 0.0f;
      for (int m = 0; m < M_; ++m) s = fmaf(wv[m], mem[m][tid], s);
      reads_buf[base + tid] = s;
    }
    __syncthreads();

    for (int i = tid; i < M_ * DV_; i += 256) {  // erase/add update
      int m = i >> 6, v = i & 63;
      float wm = wv[m];
      mem[m][v] = mem[m][v] * (1.0f - wm * ev[v]) + wm * av[v];
    }
    __syncthreads();
  }
}

// ---------------------------------------------------------------------------
// Kernel 3: x=[q|read] (BTx128) @ fc1_W (128x128) via f32 WMMA, ReLU, 128->1
// head via per-lane partials + shfl_xor reduction, sigmoid, mask; also emits
// out2 = correct*mask.
// ---------------------------------------------------------------------------
__global__ __launch_bounds__(256) void dkvmn_fc(
    const int* __restrict__ skill_seq, const int* __restrict__ correct_seq,
    const float* __restrict__ mask, const float* __restrict__ skill_embed,
    const float* __restrict__ reads_buf, const float* __restrict__ fc1_W,
    const float* __restrict__ fc1_b, const float* __restrict__ fc2_W,
    const float* __restrict__ fc2_b, float* __restrict__ out_p,
    float* __restrict__ out_c) {
  __shared__ float xt[8][16][H_];  // 64 KB

  const int tid = threadIdx.x;
  const int wave = tid >> 5;
  const int lane = tid & 31;
  const int r0 = (blockIdx.x * 8 + wave) * 16;

  {  // stage x = [q, read]
    int row = lane >> 1, half = (lane & 1) * 32;
    int g = r0 + row;
    int s = skill_seq[g];
    const float4* qsrc = (const float4*)(skill_embed + (size_t)s * DK_ + half);
    const float4* rsrc = (const float4*)(reads_buf + (size_t)g * 64 + half);
    float4* xq = (float4*)&xt[wave][row][half];
    float4* xr = (float4*)&xt[wave][row][64 + half];
#pragma unroll
    for (int j = 0; j < 8; ++j) {
      xq[j] = qsrc[j];
      xr[j] = rsrc[j];
    }
  }
  __syncthreads();

  const int mrow = lane & 15;
  const int kh = (lane >> 4) << 1;
  const int dcol = lane & 15;
  const int rbase = (lane < 16) ? 0 : 8;

  v8f acc[8];
#pragma unroll
  for (int nt = 0; nt < 8; ++nt) acc[nt] = (v8f)0.0f;
  for (int kk = 0; kk < 32; ++kk) {
    int k0 = kk * 4;
    v2f av;
    av.x = xt[wave][mrow][k0 + kh];
    av.y = xt[wave][mrow][k0 + kh + 1];
#pragma unroll
    for (int nt = 0; nt < 8; ++nt) {
      int n = nt * 16 + mrow;
      v2f bv;
      bv.x = fc1_W[(size_t)(k0 + kh) * H_ + n];
      bv.y = fc1_W[(size_t)(k0 + kh + 1) * H_ + n];
      acc[nt] = wmma_f32_k4(av, bv, acc[nt]);
    }
  }

  const float b2 = fc2_b[0];
#pragma unroll
  for (int j = 0; j < 8; ++j) {
    float s = 0.0f;
#pragma unroll
    for (int nt = 0; nt < 8; ++nt) {
      int col = nt * 16 + dcol;
      float h = acc[nt][j] + fc1_b[col];
      h = fmaxf(h, 0.0f);
      s = fmaf(h, fc2_W[col], s);
    }
    // reduce across the 16-lane row group (D layout halves of wave32)
    s += __shfl_xor(s, 1, 32);
    s += __shfl_xor(s, 2, 32);
    s += __shfl_xor(s, 4, 32);
    s += __shfl_xor(s, 8, 32);
    if (dcol == 0) {
      int g = r0 + rbase + j;
      float mk = mask[g];
      float p = 1.0f / (1.0f + __expf(-(s + b2)));
      out_p[g] = p * mk;
      out_c[g] = (float)correct_seq[g] * mk;
    }
  }
}

// ---------------------------------------------------------------------------
extern "C" void kernel_launch(void* const* d_in, const int* in_sizes, int n_in,
                              void* d_out, int out_size, void* d_ws,
                              size_t ws_size, hipStream_t stream) {
  const int* skill_seq = (const int*)d_in[0];
  const int* correct_seq = (const int*)d_in[1];
  const float* mask = (const float*)d_in[2];
  const float* skill_embed = (const float*)d_in[3];
  const float* key_memory = (const float*)d_in[4];
  const float* value_init = (const float*)d_in[5];
  const float* interaction_embed = (const float*)d_in[6];
  const float* erase_W = (const float*)d_in[7];
  const float* erase_b = (const float*)d_in[8];
  const float* add_W = (const float*)d_in[9];
  const float* add_b = (const float*)d_in[10];
  const float* fc1_W = (const float*)d_in[11];
  const float* fc1_b = (const float*)d_in[12];
  const float* fc2_W = (const float*)d_in[13];
  const float* fc2_b = (const float*)d_in[14];

  float* w_buf = (float*)d_ws;                  // BT x 64 (w padded)
  float* e_buf = w_buf + (size_t)BT_ * 64;      // BT x 64
  float* a_buf = e_buf + (size_t)BT_ * 64;      // BT x 64
  float* reads_buf = a_buf + (size_t)BT_ * 64;  // BT x 64

  float* out_p = (float*)d_out;
  float* out_c = out_p + BT_;

  dkvmn_precompute<<<BT_ / 128, 256, 0, stream>>>(
      skill_seq, correct_seq, skill_embed, key_memory, interaction_embed,
      erase_W, erase_b, add_W, add_b, w_buf, e_buf, a_buf);
  dkvmn_scan<<<B_, 256, 0, stream>>>(value_init, w_buf, e_buf, a_buf,
                                     reads_buf);
  dkvmn_fc<<<BT_ / 128, 256, 0, stream>>>(skill_seq, correct_seq, mask,
                                          skill_embed, reads_buf, fc1_W, fc1_b,
                                          fc2_W, fc2_b, out_p, out_c);
}